// Barrier_Net_10505490006166
// MI455X (gfx1250) — compile-verified
//
#include <hip/hip_runtime.h>
#include <math.h>

#define NW 8
#define BLOCK_THREADS (NW * 32)
#define AG_PER_BLOCK (NW * 16)
#define LDX 72    // padded row stride for x tile (68 cols)
#define LDA 68    // padded row stride for 16x64 activation tile (conflict-free A-frag reads)
#define LDW 160   // floats per K-pair row of swizzled 64x64 weights (160%64==32 -> halves on disjoint banks)

typedef __attribute__((ext_vector_type(2))) float v2f;
typedef __attribute__((ext_vector_type(8))) float v8f;

__device__ __forceinline__ v8f wmma4(v2f a, v2f b, v8f c) {
  // V_WMMA_F32_16X16X4_F32 : D(16x16,f32) = A(16x4,f32) x B(4x16,f32) + C
  return __builtin_amdgcn_wmma_f32_16x16x4_f32(false, a, false, b, (short)0, c,
                                               false, false);
}

__device__ __forceinline__ v8f splat8(float v) {
  v8f r = {v, v, v, v, v, v, v, v};
  return r;
}

__device__ __forceinline__ float relu(float v) {
  // single v_med3_f32: median(v, 0, +inf) == clamp to [0, inf) == relu (finite data)
  return __builtin_amdgcn_fmed3f(v, 0.0f, __builtin_inff());
}

// [16 x 64](act, row-stride LDA) @ [64 x 64](Wp, K-pair swizzled) + bscale*bias (opt. relu)
__device__ __forceinline__ void gemm_16x64x64(const float* act, const float* Wp,
                                              const float* bias, float bscale,
                                              bool dorelu, int nlo, int khalf,
                                              v8f o[4]) {
  v2f af[16];
#pragma unroll
  for (int k = 0; k < 16; ++k) {  // A fragments reused across all 4 N-tiles
    af[k].x = act[nlo * LDA + k * 4 + khalf];
    af[k].y = act[nlo * LDA + k * 4 + khalf + 1];
  }
#pragma unroll
  for (int nt = 0; nt < 4; ++nt) {
    const int col = nt * 16 + nlo;
    v8f acc = splat8(bscale * bias[col]);     // bias folded into C
#pragma unroll
    for (int k = 0; k < 16; ++k) {
      const int p = k * 2 + (khalf >> 1);     // K-pair row = (k*4 + khalf)/2
      v2f b;
      b.x = Wp[p * LDW + col * 2];            // adjacent -> single ds_load_b64
      b.y = Wp[p * LDW + col * 2 + 1];
      acc = wmma4(af[k], b, acc);
    }
    if (dorelu) {
#pragma unroll
      for (int v = 0; v < 8; ++v) acc[v] = relu(acc[v]);
    }
    o[nt] = acc;
  }
}

// [16 x 64](act) @ [64 x 2](W) + b -> one C fragment, cols 0..1 valid
__device__ __forceinline__ v8f gemm_16x64x2(const float* act, const float* W,
                                            const float* b2, int nlo, int khalf) {
  v8f acc = splat8((nlo < 2) ? b2[nlo] : 0.0f);
#pragma unroll
  for (int k = 0; k < 16; ++k) {
    v2f a, bf;
    a.x = act[nlo * LDA + k * 4 + khalf];
    a.y = act[nlo * LDA + k * 4 + khalf + 1];
    bf.x = (nlo < 2) ? W[(k * 4 + khalf) * 2 + nlo] : 0.0f;
    bf.y = (nlo < 2) ? W[(k * 4 + khalf + 1) * 2 + nlo] : 0.0f;
    acc = wmma4(a, bf, acc);
  }
  return acc;
}

// scatter C fragments (4 N-tiles) into 16x64 LDS tile (row-stride LDA)
__device__ __forceinline__ void store_tile(float* act, int nlo, int mh,
                                           const v8f c[4]) {
#pragma unroll
  for (int nt = 0; nt < 4; ++nt) {
#pragma unroll
    for (int v = 0; v < 8; ++v) {
      act[(mh + v) * LDA + nt * 16 + nlo] = c[nt][v];
    }
  }
}

__global__ __launch_bounds__(BLOCK_THREADS)
void barrier_net_kernel(const float* __restrict__ x,
                        const float* __restrict__ phiW1, const float* __restrict__ phib1,
                        const float* __restrict__ phiW2, const float* __restrict__ phib2,
                        const float* __restrict__ rhoW1, const float* __restrict__ rhob1,
                        const float* __restrict__ rhoW2, const float* __restrict__ rhob2,
                        const float* __restrict__ psiW1, const float* __restrict__ psib1,
                        const float* __restrict__ psiW2, const float* __restrict__ psib2,
                        const float* __restrict__ psiW3, const float* __restrict__ psib3,
                        float* __restrict__ out) {
  __shared__ float sPhiW1[4 * 64], sPhiB1[64], sPhiW2[32 * LDW], sPhiB2[64];
  __shared__ float sRhoW1[32 * LDW], sRhoB1[64], sRhoW2[64 * 2], sRhoB2[2];
  __shared__ float sPsiW1[4 * 64], sPsiB1[64], sPsiW2[32 * LDW], sPsiB2[64];
  __shared__ float sPsiW3[64 * 2], sPsiB3[2];
  __shared__ float sXS[NW][16 * LDX];
  __shared__ float sACT[NW][16 * LDA];
  __shared__ float sRT[NW][32];
  __shared__ float sBAR[NW][32];   // two half-wave partials per agent

  const int tid = threadIdx.x;
  // 64x64 weights: swizzle rows k,k+1 adjacent per column (K-pair layout)
  for (int i = tid; i < 64 * 64; i += BLOCK_THREADS) {
    int k = i >> 6, n = i & 63;
    int d = (k >> 1) * LDW + n * 2 + (k & 1);
    sPhiW2[d] = phiW2[i];
    sRhoW1[d] = rhoW1[i];
    sPsiW2[d] = psiW2[i];
  }
  for (int i = tid; i < 4 * 64; i += BLOCK_THREADS) {
    sPhiW1[i] = phiW1[i];
    sPsiW1[i] = psiW1[i];
  }
  for (int i = tid; i < 64; i += BLOCK_THREADS) {
    sPhiB1[i] = phib1[i];
    sPhiB2[i] = phib2[i];
    sRhoB1[i] = rhob1[i];
    sPsiB1[i] = psib1[i];
    sPsiB2[i] = psib2[i];
  }
  for (int i = tid; i < 128; i += BLOCK_THREADS) {
    sRhoW2[i] = rhoW2[i];
    sPsiW3[i] = psiW3[i];
  }
  if (tid < 2) {
    sRhoB2[tid] = rhob2[tid];
    sPsiB3[tid] = psib3[tid];
  }

  const int lane = tid & 31;
  const int wave = tid >> 5;
  const int ag0 = blockIdx.x * AG_PER_BLOCK + wave * 16;
  float* xs = sXS[wave];
  float* act = sACT[wave];
  float* rt = sRT[wave];
  float* bar = sBAR[wave];

  // stage 0: coalesced load of the wave's 16x68 input tile into LDS
  const float* xsrc = x + (size_t)ag0 * 68;
  for (int i = lane; i < 16 * 68; i += 32) {
    int m = i / 68;
    int c = i - m * 68;
    xs[m * LDX + c] = xsrc[i];
  }
  __syncthreads();

  const int nlo = lane & 15;            // column within 16-wide tile
  const int hi = lane >> 4;             // half-wave id
  const int khalf = hi << 1;            // 0 or 2 (A/B fragment K base)
  const int mh = hi << 3;               // 0 or 8 (C fragment row base)

  // ---- phi layer 1 (K=4) + relu, summed over 17 chunks ----
  v2f b1f[4];
  v8f cb1[4];
#pragma unroll
  for (int nt = 0; nt < 4; ++nt) {
    b1f[nt].x = sPhiW1[khalf * 64 + nt * 16 + nlo];
    b1f[nt].y = sPhiW1[(khalf + 1) * 64 + nt * 16 + nlo];
    cb1[nt] = splat8(sPhiB1[nt * 16 + nlo]);
  }
  v2f xf[17];  // prefetch all chunk A-fragments (hide LDS latency)
#pragma unroll
  for (int c = 0; c < 17; ++c) {
    xf[c].x = xs[nlo * LDX + c * 4 + khalf];
    xf[c].y = xs[nlo * LDX + c * 4 + khalf + 1];
  }
  v8f s1[4];
#pragma unroll
  for (int nt = 0; nt < 4; ++nt) {      // chunk 0 initializes the accumulator
    v8f h = wmma4(xf[0], b1f[nt], cb1[nt]);
#pragma unroll
    for (int v = 0; v < 8; ++v) s1[nt][v] = relu(h[v]);
  }
#pragma unroll
  for (int c = 1; c < 17; ++c) {
#pragma unroll
    for (int nt = 0; nt < 4; ++nt) {
      v8f h = wmma4(xf[c], b1f[nt], cb1[nt]);  // bias via C operand
#pragma unroll
      for (int v = 0; v < 8; ++v) s1[nt][v] += relu(h[v]);
    }
  }

  // ---- phi layer 2: y2 = s1 @ W2phi + 17*b2phi (last layer is linear) ----
  store_tile(act, nlo, mh, s1);
  __syncthreads();
  v8f y2[4];
  gemm_16x64x64(act, sPhiW2, sPhiB2, 17.0f, false, nlo, khalf, y2);

  // ---- rho layer 1 + relu ----
  store_tile(act, nlo, mh, y2);
  __syncthreads();
  v8f r1[4];
  gemm_16x64x64(act, sRhoW1, sRhoB1, 1.0f, true, nlo, khalf, r1);

  // ---- rho layer 2 (64 -> 2), stash per-agent result ----
  store_tile(act, nlo, mh, r1);
  __syncthreads();
  v8f rc = gemm_16x64x2(act, sRhoW2, sRhoB2, nlo, khalf);
  if (nlo < 2) {
#pragma unroll
    for (int v = 0; v < 8; ++v) rt[(mh + v) * 2 + nlo] = rc[v];
  }
  __syncthreads();

  // ---- psi layer 1: input = [rho_out(2), goal(2)], K=4 ----
  v2f a5;
  if (lane < 16) {  // khalf==0 : rho_out columns
    a5.x = rt[nlo * 2 + 0];
    a5.y = rt[nlo * 2 + 1];
  } else {          // khalf==2 : goal = x[:,0:2]
    a5.x = xs[nlo * LDX + 0];
    a5.y = xs[nlo * LDX + 1];
  }
  v8f p1[4];
#pragma unroll
  for (int nt = 0; nt < 4; ++nt) {
    v2f bw;
    bw.x = sPsiW1[khalf * 64 + nt * 16 + nlo];
    bw.y = sPsiW1[(khalf + 1) * 64 + nt * 16 + nlo];
    v8f h = wmma4(a5, bw, splat8(sPsiB1[nt * 16 + nlo]));
#pragma unroll
    for (int v = 0; v < 8; ++v) p1[nt][v] = relu(h[v]);
  }

  // ---- psi layer 2 + relu ----
  store_tile(act, nlo, mh, p1);
  __syncthreads();
  v8f p2[4];
  gemm_16x64x64(act, sPsiW2, sPsiB2, 1.0f, true, nlo, khalf, p2);

  // ---- psi layer 3 (64 -> 2): empty_action ----
  store_tile(act, nlo, mh, p2);
  __syncthreads();
  v8f ea = gemm_16x64x2(act, sPsiW3, sPsiB3, nlo, khalf);

  // ---- analytic barrier term: each half-wave does 8 neighbors of one agent ----
  {
    const float* row = xs + nlo * LDX;
    const int cbase = 1 + hi * 8;
    float s = 0.0f;
#pragma unroll
    for (int j = 0; j < 8; ++j) {
      const int c = cbase + j;
      float px = row[c * 4 + 0];
      float py = row[c * 4 + 1];
      float vx = row[c * 4 + 2];
      float vy = row[c * 4 + 3];
      float dist = sqrtf(px * px + py * py);  // |dp| == |p|
      // h = sqrt(4*A_MAX*(dist-DS)) + dot(dv,dp)/dist ; dv.dp == v.p
      float h = sqrtf(2.0f * (dist - 0.4f)) + (vx * px + vy * py) / dist;
      s += 1.0f / h;
    }
    bar[hi * 16 + nlo] = s;  // partial for neighbors [cbase, cbase+8)
  }
  __syncthreads();

  // ---- (tanh(ea+barrier)+1)*0.5*(A_MAX-A_MIN)+A_MIN == 0.5*tanh(..) ----
  if (nlo < 2) {
#pragma unroll
    for (int v = 0; v < 8; ++v) {
      float btot = bar[mh + v] + bar[16 + mh + v];
      float a = tanhf(ea[v] + btot);
      out[(size_t)(ag0 + mh + v) * 2 + nlo] = 0.5f * a;
    }
  }
}

extern "C" void kernel_launch(void* const* d_in, const int* in_sizes, int n_in,
                              void* d_out, int out_size, void* d_ws, size_t ws_size,
                              hipStream_t stream) {
  (void)n_in; (void)out_size; (void)d_ws; (void)ws_size;
  const float* x     = (const float*)d_in[0];
  const float* phiW1 = (const float*)d_in[1];
  const float* phib1 = (const float*)d_in[2];
  const float* phiW2 = (const float*)d_in[3];
  const float* phib2 = (const float*)d_in[4];
  const float* rhoW1 = (const float*)d_in[5];
  const float* rhob1 = (const float*)d_in[6];
  const float* rhoW2 = (const float*)d_in[7];
  const float* rhob2 = (const float*)d_in[8];
  const float* psiW1 = (const float*)d_in[9];
  const float* psib1 = (const float*)d_in[10];
  const float* psiW2 = (const float*)d_in[11];
  const float* psib2 = (const float*)d_in[12];
  const float* psiW3 = (const float*)d_in[13];
  const float* psib3 = (const float*)d_in[14];
  float* out = (float*)d_out;

  int nAgents = in_sizes[0] / 68;
  int nBlocks = nAgents / AG_PER_BLOCK;  // 131072 / 128 = 1024
  barrier_net_kernel<<<nBlocks, BLOCK_THREADS, 0, stream>>>(
      x, phiW1, phib1, phiW2, phib2, rhoW1, rhob1, rhoW2, rhob2,
      psiW1, psib1, psiW2, psib2, psiW3, psib3, out);
}